// GCNConv_9328668967304
// MI455X (gfx1250) — compile-verified
//
#include <hip/hip_runtime.h>

typedef __attribute__((ext_vector_type(2))) float v2f;
typedef __attribute__((ext_vector_type(8))) float v8f;

constexpr int N = 8192;
constexpr int E = 131072;
constexpr int NODE_DIM = 128;    // K
constexpr int HIDDEN_DIM = 256;  // output columns

// LDS W staging: pad rows 128 -> 132 floats so B-fragment ds_load_b64 is bank-conflict-free
constexpr int W_LDS_STRIDE = 132;
constexpr size_t W_LDS_BYTES = (size_t)HIDDEN_DIM * W_LDS_STRIDE * sizeof(float); // 132 KB

// ---- workspace layout (bytes) ----
constexpr size_t BITMAP_WORDS = (size_t)N * (size_t)N / 32;        // 2,097,152 u32 = 8 MB
constexpr size_t OFF_BITMAP = 0;
constexpr size_t OFF_DEG    = OFF_BITMAP + BITMAP_WORDS * 4;       // N u32
constexpr size_t OFF_DINV   = OFF_DEG + (size_t)N * 4;             // N f32
constexpr size_t OFF_FLAGS  = OFF_DINV + (size_t)N * 4;            // E u8
constexpr size_t OFF_Z      = (OFF_FLAGS + (size_t)E + 255) & ~(size_t)255; // N*128 f32 (4 MB)

// Zero the dedup bitmap with b128 stores; set deg[i] = 1 (the +I self loop).
__global__ void k_init(uint4* __restrict__ bm4, unsigned* __restrict__ deg) {
    size_t i = (size_t)blockIdx.x * blockDim.x + threadIdx.x;
    if (i < BITMAP_WORDS / 4) bm4[i] = make_uint4(0u, 0u, 0u, 0u);
    if (i < (size_t)N)        deg[i] = 1u;
}

// Dedup edges via N*N bitmap; first occurrence "owns" the edge:
// sets flags[e]=1 and increments deg[src] (row-sum over targets, axis=1).
__global__ void k_mark(const long long* __restrict__ src,
                       const long long* __restrict__ tgt,
                       unsigned* __restrict__ bitmap,
                       unsigned* __restrict__ deg,
                       unsigned char* __restrict__ flags) {
    int e = blockIdx.x * blockDim.x + threadIdx.x;
    if (e >= E) return;
    int s = (int)src[e];
    int t = (int)tgt[e];
    size_t idx = (size_t)s * N + (size_t)t;
    unsigned bit = 1u << (idx & 31u);
    unsigned old = atomicOr(&bitmap[idx >> 5], bit);
    unsigned char own = (old & bit) ? 0u : 1u;
    flags[e] = own;
    if (own) atomicAdd(&deg[s], 1u);
}

// dinv[i] = deg[i]^-0.5 ; z[i,:] = dinv[i]*x[i,:]  (the +I diagonal term), float4 wide.
__global__ void k_scale_init(const float4* __restrict__ x4,
                             const unsigned* __restrict__ deg,
                             float* __restrict__ dinv,
                             float4* __restrict__ z4) {
    int i = blockIdx.x * blockDim.x + threadIdx.x;   // over N * (NODE_DIM/4)
    int node = i >> 5;                               // 32 float4 per node
    float di = 1.0f / sqrtf((float)deg[node]);
    if ((i & 31) == 0) dinv[node] = di;
    float4 xv = x4[i];
    z4[i] = make_float4(di * xv.x, di * xv.y, di * xv.z, di * xv.w);
}

// Wave-per-edge scatter: for each owned edge (s -> t): z[s,:] += dinv[t]*x[t,:].
// Each lane handles 4 dims: one b128 load of x, 4 fp32 atomic adds into z (L2-resident).
__global__ void k_scatter(const long long* __restrict__ src,
                          const long long* __restrict__ tgt,
                          const unsigned char* __restrict__ flags,
                          const float* __restrict__ x,
                          const float* __restrict__ dinv,
                          float* __restrict__ z) {
    int e = blockIdx.x * (blockDim.x >> 5) + (threadIdx.x >> 5);
    int lane = threadIdx.x & 31;
    if (e >= E) return;
    if (!flags[e]) return;                 // wave-uniform in practice
    int s = (int)src[e];
    int t = (int)tgt[e];
    float dt = dinv[t];
    float4 xv = *(const float4*)(x + (size_t)t * NODE_DIM + lane * 4);
    float* zp = z + (size_t)s * NODE_DIM + lane * 4;
    atomicAdd(zp + 0, dt * xv.x);
    atomicAdd(zp + 1, dt * xv.y);
    atomicAdd(zp + 2, dt * xv.z);
    atomicAdd(zp + 3, dt * xv.w);
}

// out[i,h] = dinv[i] * sum_k z[i,k] * W[h,k]   via V_WMMA_F32_16X16X4_F32.
//
// Workgroup = 8 waves, covers 128 rows of z. All of W staged in LDS (padded rows).
// Each wave keeps its 16x128 A-panel in 64 VGPRs (loaded once, overlapped with W
// staging). Per n-tile: preload the whole B-panel into 64 VGPRs; sched_barrier(0)
// pins the batched ds_load_2addr_b64 block ahead of the 32-WMMA chain so only one
// LDS latency is paid per tile instead of sixteen.
//
// Fragment layouts (ISA 7.12.2, 32-bit):
//   A 16x4: lanes 0-15 -> rows, hold (K0,K1); lanes 16-31 hold (K2,K3)
//   B 4x16: lanes 0-15 -> cols, hold (K0,K1); lanes 16-31 hold (K2,K3)
//   C/D 16x16: VGPR v -> M = v + 8*(lane>=16), N = lane&15
__global__ void __launch_bounds__(256)
k_gemm_wmma(const float* __restrict__ z,
            const float* __restrict__ W,
            const float* __restrict__ dinv,
            float* __restrict__ out) {
    extern __shared__ float sW[];   // HIDDEN_DIM x W_LDS_STRIDE

    int wave = threadIdx.x >> 5;
    int lane = threadIdx.x & 31;
    int m_tile = blockIdx.x * 8 + wave;                 // 0..511 (N/16)

    int r  = lane & 15;
    int kh = lane >> 4;                                 // 0: (K0,K1), 1: (K2,K3)

    // --- A panel: load 16x128 z-tile into registers (issued before the barrier) ---
    v2f a[32];
    {
        const float* zrow = z + (size_t)(m_tile * 16 + r) * NODE_DIM + 2 * kh;
#pragma unroll
        for (int k = 0; k < 32; ++k) a[k] = *(const v2f*)(zrow + 4 * k);
    }

    // --- per-lane row scales for the 8 output rows this lane stores ---
    float dv[8];
#pragma unroll
    for (int v = 0; v < 8; ++v) dv[v] = dinv[m_tile * 16 + v + 8 * kh];

    // --- stage all of W into LDS: one row (128 floats = 32 float4) per thread ---
    {
        int row = threadIdx.x;                          // 0..255 == HIDDEN_DIM rows
        const float4* srcv = (const float4*)(W + (size_t)row * NODE_DIM);
        float4* dstv = (float4*)(sW + (size_t)row * W_LDS_STRIDE);
#pragma unroll
        for (int c = 0; c < NODE_DIM / 4; ++c) dstv[c] = srcv[c];
    }
    __syncthreads();

    // --- sweep the 16 n-tiles ---
    for (int nt = 0; nt < HIDDEN_DIM / 16; ++nt) {
        const float* wrow = sW + (size_t)(nt * 16 + r) * W_LDS_STRIDE + 2 * kh;

        // Preload full B-panel for this n-tile: batched LDS loads.
        v2f b[32];
#pragma unroll
        for (int k = 0; k < 32; ++k) b[k] = *(const v2f*)(wrow + 4 * k);

        // Hard scheduling fence: nothing crosses. Keeps all 16 ds_load_2addr_b64
        // issued back-to-back before the WMMA chain (one amortized s_wait_dscnt)
        // instead of being re-sunk next to each consumer.
        __builtin_amdgcn_sched_barrier(0);

        v8f c = {};
#pragma unroll
        for (int k = 0; k < 32; ++k) {
            // (neg_a, A, neg_b, B, c_mod, C, reuse_a, reuse_b)
            c = __builtin_amdgcn_wmma_f32_16x16x4_f32(false, a[k], false, b[k],
                                                      (short)0, c, false, false);
        }
        __builtin_amdgcn_sched_barrier(0);

        int col = nt * 16 + r;
        float* orow = out + (size_t)(m_tile * 16 + 8 * kh) * HIDDEN_DIM + col;
#pragma unroll
        for (int v = 0; v < 8; ++v) {
            orow[(size_t)v * HIDDEN_DIM] = dv[v] * c[v];
        }
    }
}

extern "C" void kernel_launch(void* const* d_in, const int* in_sizes, int n_in,
                              void* d_out, int out_size, void* d_ws, size_t ws_size,
                              hipStream_t stream) {
    const float*     x  = (const float*)d_in[0];                 // [N,128] f32
    const long long* ei = (const long long*)d_in[1];             // [2,E] i64
    const float*     W  = (const float*)d_in[2];                 // [256,128] f32
    float* out = (float*)d_out;                                  // [N,256] f32

    char* ws = (char*)d_ws;
    unsigned*      bitmap = (unsigned*)(ws + OFF_BITMAP);
    unsigned*      deg    = (unsigned*)(ws + OFF_DEG);
    float*         dinv   = (float*)(ws + OFF_DINV);
    unsigned char* flags  = (unsigned char*)(ws + OFF_FLAGS);
    float*         z      = (float*)(ws + OFF_Z);

    const long long* src = ei;        // row 0
    const long long* tgt = ei + E;    // row 1

    k_init<<<(int)(BITMAP_WORDS / 4 / 256), 256, 0, stream>>>((uint4*)bitmap, deg);
    k_mark<<<E / 256, 256, 0, stream>>>(src, tgt, bitmap, deg, flags);
    k_scale_init<<<(N * NODE_DIM / 4) / 256, 256, 0, stream>>>(
        (const float4*)x, deg, dinv, (float4*)z);
    k_scatter<<<E / 8, 256, 0, stream>>>(src, tgt, flags, x, dinv, z);
    k_gemm_wmma<<<N / 16 / 8, 256, (int)W_LDS_BYTES, stream>>>(z, W, dinv, out);
}